// NodeEncoder_69896297775693
// MI455X (gfx1250) — compile-verified
//
#include <hip/hip_runtime.h>

// ---------------------------------------------------------------------------
// GCN NodeEncoder for MI455X (gfx1250, wave32, WMMA).
// N=50000 nodes, E=600000 edges, D=128, L=3.
//
// Strategy:
//  * fp32 WMMA (v_wmma_f32_16x16x4_f32) for all 5 dense GEMMs (full precision,
//    GEMMs are HBM-bound anyway: ~256MB traffic @ 23.3TB/s ~= 11us).
//  * CSR-gather aggregation (no fp32 atomics): xw (25.6MB) lives in the 192MB
//    L2, so the ~307MB/layer of neighbor gathers are L2 reads.
//  * Deterministic: CSR rows sorted; BN reduction is two-stage, fixed order.
// ---------------------------------------------------------------------------

#define NN   50000
#define NE_  600000
#define DD   128
#define MT_  (NN / 16)        // 3125 exact M-tiles of 16
#define LDSP 66               // padded LDS stride for transposed W (bank-safe)
#define BN_BLOCKS 256

typedef float v2f __attribute__((ext_vector_type(2)));
typedef float v8f __attribute__((ext_vector_type(8)));

// ----------------------------- small utilities -----------------------------

__global__ void zero_i32_kernel(int* p, int n) {
    int i = blockIdx.x * blockDim.x + threadIdx.x;
    if (i < n) p[i] = 0;
}

__global__ void count_edges_kernel(const int* __restrict__ dst, int* cnt, int ne) {
    int i = blockIdx.x * blockDim.x + threadIdx.x;
    if (i < ne) atomicAdd(&cnt[dst[i]], 1);
}

__global__ void dinv_kernel(const int* __restrict__ cnt, float* dinv, int n) {
    int i = blockIdx.x * blockDim.x + threadIdx.x;
    if (i < n) dinv[i] = rsqrtf((float)cnt[i] + 1.0f);   // deg includes self-loop
}

// Single-block exclusive scan (Hillis-Steele over 1024-wide tiles).
__global__ __launch_bounds__(1024) void scan_excl_kernel(const int* __restrict__ cnt,
                                                         int* row_ptr, int n) {
    __shared__ int sdata[1024];
    __shared__ int s_running;
    const int tid = threadIdx.x;
    if (tid == 0) s_running = 0;
    __syncthreads();
    for (int base = 0; base < n; base += 1024) {
        int i = base + tid;
        int v = (i < n) ? cnt[i] : 0;
        sdata[tid] = v;
        __syncthreads();
        for (int off = 1; off < 1024; off <<= 1) {
            int t2 = (tid >= off) ? sdata[tid - off] : 0;
            __syncthreads();
            sdata[tid] += t2;
            __syncthreads();
        }
        int incl = sdata[tid];
        int offset = s_running;
        if (i < n) row_ptr[i] = offset + incl - v;
        __syncthreads();
        if (tid == 1023) s_running = offset + sdata[1023];
        __syncthreads();
    }
    if (threadIdx.x == 0) row_ptr[n] = s_running;
}

__global__ void fill_csr_kernel(const int* __restrict__ src, const int* __restrict__ dst,
                                const int* __restrict__ row_ptr, int* cursor,
                                int* col_idx, int ne) {
    int i = blockIdx.x * blockDim.x + threadIdx.x;
    if (i >= ne) return;
    int d = dst[i];
    int pos = atomicAdd(&cursor[d], 1);
    col_idx[row_ptr[d] + pos] = src[i];
}

// Insertion-sort each CSR row (avg degree ~12) -> deterministic sum order.
__global__ void sort_rows_kernel(int* col_idx, const int* __restrict__ row_ptr, int n) {
    int i = blockIdx.x * blockDim.x + threadIdx.x;
    if (i >= n) return;
    int b = row_ptr[i], e = row_ptr[i + 1];
    for (int j = b + 1; j < e; ++j) {
        int key = col_idx[j];
        int k = j - 1;
        while (k >= b && col_idx[k] > key) { col_idx[k + 1] = col_idx[k]; --k; }
        col_idx[k + 1] = key;
    }
}

// ------------------------------- WMMA GEMM ---------------------------------
// C[M,128] = A[M,128] @ W[128,128] (+bias) (+relu).  One wave per 16-row tile;
// 8 N-tiles of 16 held in 8 v8f accumulators; K split in 2 halves of 64 so the
// transposed W-copy in LDS stays at 33KB.  fp32 WMMA shape is 16x16x4.
// A frag (lane L): row = Mtile*16 + (L&15), K pair = {kb, kb+1}, kb = 4k + 2*(L>>4).
// B frag (lane L): col = Ntile*16 + (L&15), same K pair  -> contiguous float2
//                  reads from transposed W in LDS.
// D (vgpr r, lane L): row = Mtile*16 + r + 8*(L>>4), col = Ntile*16 + (L&15).
__global__ __launch_bounds__(128) void gemm_wmma_kernel(const float* A,
                                                        const float* __restrict__ W,
                                                        const float* __restrict__ bias,
                                                        float* C,
                                                        int mtiles, int do_relu) {
    __shared__ float wT[128 * LDSP];   // [col][k-in-half], padded stride
    const int t = threadIdx.x;
    const int wave = t >> 5;
    const int lane = t & 31;
    const int l15 = lane & 15;
    const int khalf = (lane >> 4) << 1;     // 0 or 2
    const int mtile = blockIdx.x * 4 + wave;
    const bool active = (mtile < mtiles);
    const int mt = active ? mtile : 0;
    const float* arow = A + (size_t)(mt * 16 + l15) * DD;

    v8f acc[8] = {};

    for (int kk = 0; kk < 2; ++kk) {
        __syncthreads();   // protect LDS reuse between halves
        // cooperative transposed load of W rows [kk*64, kk*64+64)
        for (int i = t; i < 64 * 128; i += 128) {
            int k = i >> 7;        // 0..63
            int n = i & 127;       // 0..127 (== t -> coalesced)
            wT[n * LDSP + k] = W[(kk * 64 + k) * DD + n];
        }
        __syncthreads();
        if (active) {
            const float* ak = arow + kk * 64;
            #pragma unroll 4
            for (int k4 = 0; k4 < 16; ++k4) {
                int kb = k4 * 4 + khalf;
                v2f a = *(const v2f*)(ak + kb);
                #pragma unroll
                for (int nt = 0; nt < 8; ++nt) {
                    v2f b = *(const v2f*)(&wT[(nt * 16 + l15) * LDSP + kb]);
                    acc[nt] = __builtin_amdgcn_wmma_f32_16x16x4_f32(
                        false, a, false, b, (short)0, acc[nt], false, false);
                }
            }
        }
    }
    if (!active) return;

    const int rowoff = (lane >> 4) * 8;    // 0 or 8
    #pragma unroll
    for (int nt = 0; nt < 8; ++nt) {
        int col = nt * 16 + l15;
        float bv = bias ? bias[col] : 0.0f;
        float* crow = C + (size_t)(mtile * 16 + rowoff) * DD + col;
        #pragma unroll
        for (int r = 0; r < 8; ++r) {
            float v = acc[nt][r] + bv;
            if (do_relu) v = fmaxf(v, 0.0f);
            crow[(size_t)r * DD] = v;
        }
    }
}

// --------------------------- CSR gather aggregate ---------------------------
// agg[n] = dinv[n] * sum_{s in nbr(n)} dinv[s]*xw[s]  +  dinv[n]^2 * xw[n] + b
// One wave per node, lane = float4 chunk of the 128-wide feature row.
__global__ __launch_bounds__(256) void gather_agg_kernel(const float* __restrict__ xw,
                                                         const int* __restrict__ row_ptr,
                                                         const int* __restrict__ col_idx,
                                                         const float* __restrict__ dinv,
                                                         const float* __restrict__ bconv,
                                                         float* __restrict__ agg, int n) {
    int node = blockIdx.x * 8 + (threadIdx.x >> 5);
    if (node >= n) return;
    int lane = threadIdx.x & 31;
    const float4* xw4 = (const float4*)xw;
    float4 acc = make_float4(0.f, 0.f, 0.f, 0.f);
    int b = row_ptr[node], e = row_ptr[node + 1];
    for (int j = b; j < e; ++j) {
        int s = col_idx[j];
        float c = dinv[s];
        float4 v = xw4[(size_t)s * 32 + lane];
        acc.x += v.x * c; acc.y += v.y * c; acc.z += v.z * c; acc.w += v.w * c;
    }
    float dn = dinv[node];
    float sc = dn * dn;
    float4 sv = xw4[(size_t)node * 32 + lane];
    float4 bb = ((const float4*)bconv)[lane];
    float4 o;
    o.x = acc.x * dn + sv.x * sc + bb.x;
    o.y = acc.y * dn + sv.y * sc + bb.y;
    o.z = acc.z * dn + sv.z * sc + bb.z;
    o.w = acc.w * dn + sv.w * sc + bb.w;
    ((float4*)agg)[(size_t)node * 32 + lane] = o;
}

// ------------------------------- BatchNorm ----------------------------------

// Stage 1: per-block partial sum / sumsq per feature (no atomics, deterministic)
__global__ __launch_bounds__(128) void bn_stats_kernel(const float* __restrict__ agg,
                                                       float* __restrict__ partial, int n) {
    int d = threadIdx.x;       // feature 0..127
    float s = 0.f, s2 = 0.f;
    for (int r = blockIdx.x; r < n; r += BN_BLOCKS) {
        float v = agg[(size_t)r * DD + d];
        s += v; s2 += v * v;
    }
    partial[(size_t)blockIdx.x * 256 + d] = s;
    partial[(size_t)blockIdx.x * 256 + 128 + d] = s2;
}

// Stage 2: fixed-order final reduce; store scale/shift for the apply pass.
__global__ __launch_bounds__(128) void bn_finalize_kernel(const float* __restrict__ partial,
                                                          const float* __restrict__ gamma,
                                                          const float* __restrict__ beta,
                                                          float* __restrict__ ss) {
    int d = threadIdx.x;
    float s = 0.f, s2 = 0.f;
    for (int b = 0; b < BN_BLOCKS; ++b) {
        s  += partial[(size_t)b * 256 + d];
        s2 += partial[(size_t)b * 256 + 128 + d];
    }
    const float invN = 1.0f / (float)NN;
    float mean = s * invN;
    float var = fmaxf(s2 * invN - mean * mean, 0.0f);
    float rs = rsqrtf(var + 1e-5f);
    float scale = gamma[d] * rs;
    ss[d] = scale;
    ss[128 + d] = beta[d] - mean * scale;
}

// hnew = relu(agg*scale + shift) + h_in   (float4, in-place on agg is safe)
__global__ __launch_bounds__(256) void bn_apply_kernel(const float* __restrict__ agg,
                                                       const float* __restrict__ hin,
                                                       const float* __restrict__ ss,
                                                       float* hout, int n4) {
    int i = blockIdx.x * blockDim.x + threadIdx.x;
    if (i >= n4) return;
    int c4 = i & 31;
    const float4* ss4 = (const float4*)ss;
    float4 sc = ss4[c4];
    float4 sh = ss4[32 + c4];
    float4 a = ((const float4*)agg)[i];
    float4 h = ((const float4*)hin)[i];
    float4 o;
    o.x = fmaxf(a.x * sc.x + sh.x, 0.f) + h.x;
    o.y = fmaxf(a.y * sc.y + sh.y, 0.f) + h.y;
    o.z = fmaxf(a.z * sc.z + sh.z, 0.f) + h.z;
    o.w = fmaxf(a.w * sc.w + sh.w, 0.f) + h.w;
    ((float4*)hout)[i] = o;
}

// ------------------------------- launcher -----------------------------------

extern "C" void kernel_launch(void* const* d_in, const int* in_sizes, int n_in,
                              void* d_out, int out_size, void* d_ws, size_t ws_size,
                              hipStream_t stream) {
    (void)in_sizes; (void)n_in; (void)out_size; (void)ws_size;
    const float* x      = (const float*)d_in[0];
    const int*   edges  = (const int*)d_in[1];
    const float* w_in   = (const float*)d_in[2];
    const float* b_in   = (const float*)d_in[3];
    const float* w_conv = (const float*)d_in[4];
    const float* b_conv = (const float*)d_in[5];
    const float* gamma  = (const float*)d_in[6];
    const float* beta   = (const float*)d_in[7];
    const float* w_proj = (const float*)d_in[8];
    const float* b_proj = (const float*)d_in[9];
    float* out = (float*)d_out;

    const int* src = edges;          // edge_index[0]
    const int* dst = edges + NE_;    // edge_index[1]

    // workspace carve-out (~55MB)
    char* p = (char*)d_ws;
    auto alloc = [&](size_t bytes) {
        char* r = p; p += (bytes + 255) & ~(size_t)255; return r;
    };
    float* W0   = (float*)alloc((size_t)NN * DD * sizeof(float));
    float* W1   = (float*)alloc((size_t)NN * DD * sizeof(float));
    int*   cnt  = (int*)  alloc((size_t)NN * sizeof(int));
    int*   rp   = (int*)  alloc((size_t)(NN + 1) * sizeof(int));
    int*   ci   = (int*)  alloc((size_t)NE_ * sizeof(int));
    float* dinv = (float*)alloc((size_t)NN * sizeof(float));
    float* ss   = (float*)alloc(256 * sizeof(float));
    float* part = (float*)alloc((size_t)BN_BLOCKS * 256 * sizeof(float));

    // ---- CSR build (runs every call; deterministic result) ----
    zero_i32_kernel   <<<(NN + 255) / 256, 256, 0, stream>>>(cnt, NN);
    count_edges_kernel<<<(NE_ + 255) / 256, 256, 0, stream>>>(dst, cnt, NE_);
    dinv_kernel       <<<(NN + 255) / 256, 256, 0, stream>>>(cnt, dinv, NN);
    scan_excl_kernel  <<<1, 1024, 0, stream>>>(cnt, rp, NN);
    zero_i32_kernel   <<<(NN + 255) / 256, 256, 0, stream>>>(cnt, NN);   // reuse as cursor
    fill_csr_kernel   <<<(NE_ + 255) / 256, 256, 0, stream>>>(src, dst, rp, cnt, ci, NE_);
    sort_rows_kernel  <<<(NN + 255) / 256, 256, 0, stream>>>(ci, rp, NN);

    const int GB = (MT_ + 3) / 4;   // 4 waves (M-tiles) per 128-thread block

    // ---- input MLP: h = relu(x @ w_in + b_in) ----
    gemm_wmma_kernel<<<GB, 128, 0, stream>>>(x, w_in, b_in, W0, MT_, 1);

    float* h   = W0;
    float* xw  = W1;
    float* agg = out;   // d_out doubles as the third big buffer (fully rewritten)
    for (int l = 0; l < 3; ++l) {
        gemm_wmma_kernel<<<GB, 128, 0, stream>>>(h, w_conv + (size_t)l * DD * DD,
                                                 nullptr, xw, MT_, 0);
        gather_agg_kernel<<<(NN + 7) / 8, 256, 0, stream>>>(xw, rp, ci, dinv,
                                                            b_conv + (size_t)l * DD, agg, NN);
        bn_stats_kernel   <<<BN_BLOCKS, 128, 0, stream>>>(agg, part, NN);
        bn_finalize_kernel<<<1, 128, 0, stream>>>(part, gamma + (size_t)l * DD,
                                                  beta + (size_t)l * DD, ss);
        float* hnew = (l == 2) ? agg : xw;   // last layer: produce h in d_out
        bn_apply_kernel<<<((NN * DD / 4) + 255) / 256, 256, 0, stream>>>(
            agg, h, ss, hnew, NN * DD / 4);
        if (l < 2) { float* t = h; h = hnew; xw = t; }
        else       { h = agg; }
    }

    // ---- projection head: out = relu(h @ w_proj + b_proj), safely in-place ----
    gemm_wmma_kernel<<<GB, 128, 0, stream>>>(h, w_proj, b_proj, out, MT_, 1);
}